// MinimalRNN_55722905699022
// MI455X (gfx1250) — compile-verified
//
#include <hip/hip_runtime.h>
#include <cstdint>

// Problem constants (match reference: B=16, T=4096, H=1024)
#define B_  16
#define T_  4096
#define H_  1024
#define TC  32              // timesteps per chunk
#define NC  (T_ / TC)       // 128 chunks
#define HT  256             // channels per block tile
#define NBH (H_ / HT)       // 4 h-tiles

#define NEG_INF (-__builtin_inff())

// Pointer types matching the async-load builtin's signature:
// param0: int4 addrspace(1)* (global), param1: int4 addrspace(3)* (LDS)
typedef int v4i __attribute__((vector_size(16)));
typedef v4i __attribute__((address_space(1)))* g4p_t;
typedef v4i __attribute__((address_space(3)))* l4p_t;

// ---------------------------------------------------------------------------
// Async global -> LDS tile staging (CDNA5 ASYNCcnt path), with fallback.
// Copies TC rows x HT floats. g: row 0 / col 0 of tile (row stride = grs
// floats). l: LDS base, row stride = HT floats. 256 threads cooperate.
// ---------------------------------------------------------------------------
__device__ __forceinline__ void stage_tile(const float* __restrict__ g,
                                           float* l, int tid, int grs) {
#if __has_builtin(__builtin_amdgcn_global_load_async_to_lds_b128)
#pragma unroll
  for (int p = 0; p < (TC * HT) / (4 * 256); ++p) {   // 8 b128 ops / thread
    int q  = tid + p * 256;
    int j  = q >> 6;        // row (0..TC-1)
    int c4 = q & 63;        // float4 column
    const float* gp = g + (long)j * grs + c4 * 4;
    float*       lp = l + j * HT + c4 * 4;
    // AS1 pointer value == flat address; AS3 pointer == low 32 bits (LDS offset)
    __builtin_amdgcn_global_load_async_to_lds_b128(
        (g4p_t)(uintptr_t)gp,
        (l4p_t)(unsigned)(uintptr_t)lp,
        /*imm offset=*/0, /*cpol=*/0);
  }
#else
#pragma unroll
  for (int p = 0; p < (TC * HT) / (4 * 256); ++p) {
    int q  = tid + p * 256;
    int j  = q >> 6;
    int c4 = q & 63;
    const float4 v = *(const float4*)(g + (long)j * grs + c4 * 4);
    *(float4*)(l + j * HT + c4 * 4) = v;
  }
#endif
}

__device__ __forceinline__ void wait_async0() {
#if __has_builtin(__builtin_amdgcn_s_wait_asynccnt)
  __builtin_amdgcn_s_wait_asynccnt(0);
#else
  asm volatile("s_wait_asynccnt 0" ::: "memory");
#endif
}

// ---------------------------------------------------------------------------
// Pass 1: per-chunk carries.
//   A = sum of log_coeffs over chunk
//   L = logsumexp_i( log_values[t0+1+i] - inclusive_cumsum_i(log_coeffs) )
// Carry layout: [c][b][h]  (coalesced reads in pass 2)
// ---------------------------------------------------------------------------
__global__ __launch_bounds__(256) void k_carry(const float* __restrict__ lc,
                                               const float* __restrict__ lv,
                                               float* __restrict__ Ac,
                                               float* __restrict__ Lc) {
  __shared__ float s_lc[TC * HT];
  __shared__ float s_lv[TC * HT];
  const int c   = blockIdx.x;
  const int h0  = blockIdx.y * HT;
  const int b   = blockIdx.z;
  const int tid = threadIdx.x;
  const int t0  = c * TC;

  const float* gc = lc + ((long)b * T_ + t0) * H_ + h0;
  const float* gv = lv + ((long)b * (T_ + 1) + t0 + 1) * H_ + h0;
  stage_tile(gc, s_lc, tid, H_);
  stage_tile(gv, s_lv, tid, H_);
  wait_async0();
  __syncthreads();

  float a = 0.f, m = NEG_INF, s = 0.f;
#pragma unroll
  for (int j = 0; j < TC; ++j) {
    a += s_lc[j * HT + tid];
    float x  = s_lv[j * HT + tid] - a;
    float nm = fmaxf(m, x);
    s = s * __expf(m - nm) + __expf(x - nm);
    m = nm;
  }
  const long o = ((long)c * B_ + b) * H_ + h0 + tid;
  Ac[o] = a;
  Lc[o] = m + __logf(s);
}

// ---------------------------------------------------------------------------
// Pass 2: sequential scan over NC chunk carries per channel.
//   G_0 = log_values[b,0,h] (= log h_0);  G_{c+1} = A_c + logaddexp(G_c, L_c)
// Writes G[c][b][h] = carried log-h at each chunk start. 16K threads, tiny.
// ---------------------------------------------------------------------------
__global__ __launch_bounds__(256) void k_chunkscan(const float* __restrict__ lv,
                                                   const float* __restrict__ Ac,
                                                   const float* __restrict__ Lc,
                                                   float* __restrict__ G) {
  const int idx = blockIdx.x * 256 + threadIdx.x;  // b*H + h
  const int b = idx / H_;
  const int h = idx - b * H_;
  float g = lv[(long)b * (T_ + 1) * H_ + h];       // log h_0
  for (int c = 0; c < NC; ++c) {
    const long o = ((long)c * B_ + b) * H_ + h;
    G[o] = g;
    const float A = Ac[o];
    const float L = Lc[o];
    const float hi = fmaxf(g, L);
    g = A + hi + __logf(__expf(g - hi) + __expf(L - hi));
  }
}

// ---------------------------------------------------------------------------
// Pass 3: re-scan each chunk with incoming G and emit h_t = exp(log h_t).
//   h = exp(a + G) + exp(a + m) * s     (all exponents bounded by ~±40)
// ---------------------------------------------------------------------------
__global__ __launch_bounds__(256) void k_apply(const float* __restrict__ lc,
                                               const float* __restrict__ lv,
                                               const float* __restrict__ G,
                                               float* __restrict__ out) {
  __shared__ float s_lc[TC * HT];
  __shared__ float s_lv[TC * HT];
  const int c   = blockIdx.x;
  const int h0  = blockIdx.y * HT;
  const int b   = blockIdx.z;
  const int tid = threadIdx.x;
  const int t0  = c * TC;

  const float* gc = lc + ((long)b * T_ + t0) * H_ + h0;
  const float* gv = lv + ((long)b * (T_ + 1) + t0 + 1) * H_ + h0;
  stage_tile(gc, s_lc, tid, H_);
  stage_tile(gv, s_lv, tid, H_);
  const float gval = G[((long)c * B_ + b) * H_ + h0 + tid];
  wait_async0();
  __syncthreads();

  float a = 0.f, m = NEG_INF, s = 0.f;
  float* op = out + ((long)b * T_ + t0) * H_ + h0 + tid;
#pragma unroll
  for (int j = 0; j < TC; ++j) {
    a += s_lc[j * HT + tid];
    float x  = s_lv[j * HT + tid] - a;
    float nm = fmaxf(m, x);
    s = s * __expf(m - nm) + __expf(x - nm);
    m = nm;
    const float r = __expf(a + gval) + __expf(a + m) * s;
    __builtin_nontemporal_store(r, op + (long)j * H_);  // output never re-read
  }
}

// ---------------------------------------------------------------------------
// Launch: inputs are [0]=log_coeffs (B,T,H) f32, [1]=log_values (B,T+1,H) f32.
// Workspace: Ac, Lc, G — each NC*B*H floats (8 MB) => 24 MB total.
// ---------------------------------------------------------------------------
extern "C" void kernel_launch(void* const* d_in, const int* in_sizes, int n_in,
                              void* d_out, int out_size, void* d_ws, size_t ws_size,
                              hipStream_t stream) {
  const float* lc = (const float*)d_in[0];
  const float* lv = (const float*)d_in[1];
  float* out = (float*)d_out;

  const size_t carryN = (size_t)NC * B_ * H_;   // 2,097,152
  float* Ac = (float*)d_ws;
  float* Lc = Ac + carryN;
  float* G  = Lc + carryN;

  dim3 grid(NC, NBH, B_);   // 128 x 4 x 16 = 8192 blocks
  k_carry<<<grid, 256, 0, stream>>>(lc, lv, Ac, Lc);
  k_chunkscan<<<(B_ * H_) / 256, 256, 0, stream>>>(lv, Ac, Lc, G);
  k_apply<<<grid, 256, 0, stream>>>(lc, lv, G, out);
}